// LMFAdapter_32040456028380
// MI455X (gfx1250) — compile-verified
//
#include <hip/hip_runtime.h>
#include <math.h>

typedef __attribute__((ext_vector_type(2))) float v2f;
typedef __attribute__((ext_vector_type(8))) float v8f;

#define C_IN   384
#define K_TOT  1152
#define D_OUT  768
#define HW     1024
#define MB     64      // output rows per block
#define KC     96      // K-chunk (channels per phase)
#define APITCH 100     // LDS pitch (dwords) -> conflict-free b64 A-frag loads
#define NWAVES 16
#define LN_EPS 1e-5f

// ---------------------------------------------------------------------------
// wp [D=768][K=1152] row-major -> wpP packed [K/4][D][4]:
//   wpP[(k>>2)*D*4 + n*4 + (k&3)] = wp[n*K + k]
// so a WMMA B-fragment lane reads its two K-adjacent values with ONE b64 load,
// and a full wave's fragment covers exactly 256 contiguous bytes.
// ---------------------------------------------------------------------------
__global__ __launch_bounds__(256) void pack_wp(const float* __restrict__ wp,
                                               float* __restrict__ wpP) {
    __shared__ float t[32][33];
    const int kx = blockIdx.x * 32 + threadIdx.x;   // K coordinate
    const int n0 = blockIdx.y * 32;                 // D coordinate base
#pragma unroll
    for (int i = threadIdx.y; i < 32; i += 8)
        t[i][threadIdx.x] = wp[(size_t)(n0 + i) * K_TOT + kx];   // t[n_loc][k_loc]
    __syncthreads();
    const int n  = n0 + threadIdx.x;
    const int k0 = blockIdx.x * 32;
    {
        const int g = threadIdx.y;                  // k-group of 4 within tile (0..7)
        float4 v;
        v.x = t[threadIdx.x][4 * g + 0];
        v.y = t[threadIdx.x][4 * g + 1];
        v.z = t[threadIdx.x][4 * g + 2];
        v.w = t[threadIdx.x][4 * g + 3];
        *(float4*)&wpP[(size_t)(k0 / 4 + g) * (D_OUT * 4) + n * 4] = v;
    }
}

// ---------------------------------------------------------------------------
// Fused: dwconv3 + dwconv5 + concat + GEMM(WMMA f32) + bias + LayerNorm + GELU
// One block = 64 rows (2 image rows of one batch image) x all 768 columns.
// ---------------------------------------------------------------------------
__global__ __launch_bounds__(512) void fused_lmf_kernel(
    const float* __restrict__ x,
    const float* __restrict__ w3, const float* __restrict__ b3,
    const float* __restrict__ w5, const float* __restrict__ b5,
    const float* __restrict__ wpP, const float* __restrict__ bp,
    const float* __restrict__ gamma, const float* __restrict__ beta,
    float* __restrict__ out)
{
    __shared__ float As[MB * APITCH];     // 25.6 KB activation K-chunk
    __shared__ float redS[MB * 17];       // cross-wave sum partials
    __shared__ float redQ[MB * 17];       // cross-wave sum-of-squares partials
    __shared__ float stats[MB * 2];       // per-row mean / rstd

    const int tid = threadIdx.x;
    const int wv  = tid >> 5;             // wave id (0..15)
    const int ln  = tid & 31;
    const int lh  = ln & 15;
    const int hi  = ln >> 4;              // half-wave select
    const int m0  = blockIdx.x * MB;
    const int bimg = m0 >> 10;
    const int hw0  = m0 & 1023;
    const int h0   = hw0 >> 5;
    const float* xb = x + (size_t)bimg * C_IN * HW;

    v8f acc[4][3];
#pragma unroll
    for (int ms = 0; ms < 4; ++ms)
#pragma unroll
        for (int jn = 0; jn < 3; ++jn)
            acc[ms][jn] = v8f{0.f,0.f,0.f,0.f,0.f,0.f,0.f,0.f};

    for (int ki = 0; ki < 12; ++ki) {
        __syncthreads();                  // previous chunk's reads done
        const int sec = ki >> 2;          // 0:x  1:conv3  2:conv5
        const int c0  = (ki & 3) * KC;

        if (sec == 0) {
#pragma unroll
            for (int it = 0; it < (MB * KC) / 512; ++it) {
                const int i = tid + it * 512;
                const int row = i & 63, kk = i >> 6;
                As[row * APITCH + kk] = xb[(c0 + kk) * HW + hw0 + row];
            }
        } else if (sec == 1) {
            for (int it = 0; it < (MB * KC) / 512; ++it) {
                const int i = tid + it * 512;
                const int row = i & 63, kk = i >> 6;
                const int c = c0 + kk;
                const int h = h0 + (row >> 5), w = row & 31;
                const float* xc = xb + c * HW;
                const float* wc = w3 + c * 9;
                float s = b3[c];
#pragma unroll
                for (int dy = 0; dy < 3; ++dy) {
                    const int yy = h + dy - 1;
                    if (yy < 0 || yy > 31) continue;
#pragma unroll
                    for (int dx = 0; dx < 3; ++dx) {
                        const int xx = w + dx - 1;
                        if (xx < 0 || xx > 31) continue;
                        s = fmaf(xc[yy * 32 + xx], wc[dy * 3 + dx], s);
                    }
                }
                As[row * APITCH + kk] = s;
            }
        } else {
            for (int it = 0; it < (MB * KC) / 512; ++it) {
                const int i = tid + it * 512;
                const int row = i & 63, kk = i >> 6;
                const int c = c0 + kk;
                const int h = h0 + (row >> 5), w = row & 31;
                const float* xc = xb + c * HW;
                const float* wc = w5 + c * 25;
                float s = b5[c];
#pragma unroll
                for (int dy = 0; dy < 5; ++dy) {
                    const int yy = h + dy - 2;
                    if (yy < 0 || yy > 31) continue;
#pragma unroll
                    for (int dx = 0; dx < 5; ++dx) {
                        const int xx = w + dx - 2;
                        if (xx < 0 || xx > 31) continue;
                        s = fmaf(xc[yy * 32 + xx], wc[dy * 5 + dx], s);
                    }
                }
                As[row * APITCH + kk] = s;
            }
        }
        __syncthreads();

        const int kg4base = (ki * KC) >> 2;   // packed k-group base for this chunk
#pragma unroll 2
        for (int k4 = 0; k4 < KC / 4; ++k4) {
            const int k0 = k4 * 4;
            // A fragments (16x4 f32): lanes0-15 K={k0,k0+1}, lanes16-31 K={k0+2,k0+3}
            v2f afr[4];
#pragma unroll
            for (int ms = 0; ms < 4; ++ms)
                afr[ms] = *(const v2f*)&As[(ms * 16 + lh) * APITCH + k0 + hi * 2];
            // B fragments (4x16 f32): one coalesced b64 per lane from packed wp
            const size_t bbase = (size_t)(kg4base + k4) * (D_OUT * 4);
            v2f bfr[3];
#pragma unroll
            for (int jn = 0; jn < 3; ++jn) {
                const int col = (wv * 3 + jn) * 16 + lh;
                bfr[jn] = *(const v2f*)&wpP[bbase + col * 4 + hi * 2];
            }
#pragma unroll
            for (int ms = 0; ms < 4; ++ms)
#pragma unroll
                for (int jn = 0; jn < 3; ++jn)
                    acc[ms][jn] = __builtin_amdgcn_wmma_f32_16x16x4_f32(
                        false, afr[ms], false, bfr[jn],
                        (short)0, acc[ms][jn], false, false);
        }
    }

    // ---- epilogue: bias, LayerNorm stats (deterministic), normalize, GELU ----
#pragma unroll
    for (int jn = 0; jn < 3; ++jn) {
        const float bpv = bp[(wv * 3 + jn) * 16 + lh];
#pragma unroll
        for (int ms = 0; ms < 4; ++ms)
#pragma unroll
            for (int r = 0; r < 8; ++r)
                acc[ms][jn][r] += bpv;
    }

    // per-wave partial row sums (48 columns), reduced within 16-lane halves
#pragma unroll
    for (int ms = 0; ms < 4; ++ms) {
#pragma unroll
        for (int r = 0; r < 8; ++r) {
            float s = 0.f, q = 0.f;
#pragma unroll
            for (int jn = 0; jn < 3; ++jn) {
                const float v = acc[ms][jn][r];
                s += v;
                q = fmaf(v, v, q);
            }
#pragma unroll
            for (int msk = 1; msk < 16; msk <<= 1) {
                s += __shfl_xor(s, msk);
                q += __shfl_xor(q, msk);
            }
            const int row = ms * 16 + r + hi * 8;
            if (lh == 0) { redS[row * 17 + wv] = s; redQ[row * 17 + wv] = q; }
        }
    }
    __syncthreads();
    if (tid < MB) {
        float s = 0.f, q = 0.f;
#pragma unroll
        for (int w2 = 0; w2 < NWAVES; ++w2) { s += redS[tid * 17 + w2]; q += redQ[tid * 17 + w2]; }
        const float mean = s * (1.0f / (float)D_OUT);
        const float var  = q * (1.0f / (float)D_OUT) - mean * mean;
        stats[tid * 2]     = mean;
        stats[tid * 2 + 1] = rsqrtf(var + LN_EPS);
    }
    __syncthreads();

    float gv[3], bv[3];
#pragma unroll
    for (int jn = 0; jn < 3; ++jn) {
        const int col = (wv * 3 + jn) * 16 + lh;
        gv[jn] = gamma[col];
        bv[jn] = beta[col];
    }
#pragma unroll
    for (int ms = 0; ms < 4; ++ms) {
#pragma unroll
        for (int r = 0; r < 8; ++r) {
            const int row = ms * 16 + r + hi * 8;
            const float mean = stats[row * 2];
            const float rstd = stats[row * 2 + 1];
            const size_t ob = (size_t)(m0 + row) * D_OUT;
#pragma unroll
            for (int jn = 0; jn < 3; ++jn) {
                const int col = (wv * 3 + jn) * 16 + lh;
                const float v = acc[ms][jn][r];
                const float y = (v - mean) * rstd * gv[jn] + bv[jn];
                out[ob + col] = 0.5f * y * (1.0f + erff(y * 0.70710678118654752f));
            }
        }
    }
}

extern "C" void kernel_launch(void* const* d_in, const int* in_sizes, int n_in,
                              void* d_out, int out_size, void* d_ws, size_t ws_size,
                              hipStream_t stream) {
    const float* x     = (const float*)d_in[0];
    const float* w3    = (const float*)d_in[1];
    const float* b3    = (const float*)d_in[2];
    const float* w5    = (const float*)d_in[3];
    const float* b5    = (const float*)d_in[4];
    const float* wp    = (const float*)d_in[5];
    const float* bp    = (const float*)d_in[6];
    const float* gamma = (const float*)d_in[7];
    const float* beta  = (const float*)d_in[8];
    float* wpP = (float*)d_ws;                 // 1152*768*4 = 3.54 MB scratch
    float* out = (float*)d_out;

    dim3 tb(32, 8), tg(K_TOT / 32, D_OUT / 32);
    pack_wp<<<tg, tb, 0, stream>>>(wp, wpP);

    const int nblocks = (32 * HW) / MB;        // 512 blocks of 64 rows
    fused_lmf_kernel<<<nblocks, 512, 0, stream>>>(x, w3, b3, w5, b5, wpP, bp,
                                                  gamma, beta, out);
}